// NMF_62466004353060
// MI455X (gfx1250) — compile-verified
//
#include <hip/hip_runtime.h>
#include <math.h>

typedef __attribute__((ext_vector_type(2))) float v2f;
typedef __attribute__((ext_vector_type(8))) float v8f;

#define N_ITERS 5
#define EPS_F   1e-6f
#define ETA_F   0.02f   // 0.1 / ITERS

// ---- dynamic LDS layout (float offsets) ----
#define OM2_O 0         // omega_2 [16i][16t][64l]  (l padded 60->64)
#define OM1_O 16384     // omega_1 [16i][16t][48k]  (k padded 36->48)
#define XS_O  28672     // X slabs [16t][16i][16d]
#define YS_O  32768     // Y slabs [16i][16t][16d]
#define CM_O  36864     // cap mask [16t][64l]
#define IM_O  37888     // img mask [16i][48k]
#define SC_O  38656     // per-pair 1/(sum+eps) [256]
#define LDS_FLOATS 38912

static __device__ __forceinline__ v8f wmma4(v2f a, v2f b, v8f c) {
  // V_WMMA_F32_16X16X4_F32 : D = A(16x4) x B(4x16) + C, full fp32
  return __builtin_amdgcn_wmma_f32_16x16x4_f32(false, a, false, b, (short)0, c,
                                               false, false);
}

// ---------------- kernel 1: row-wise L2 normalization ----------------
// rows 0..4607 = imgs (128*36), rows 4608..12287 = caps (128*60); D=1024
__global__ __launch_bounds__(256) void l2norm_rows(const float* __restrict__ imgs,
                                                   const float* __restrict__ caps,
                                                   float* __restrict__ wimgs,
                                                   float* __restrict__ wcaps) {
  const int row = blockIdx.x;
  const float* src;
  float* dst;
  if (row < 128 * 36) {
    src = imgs + (size_t)row * 1024;
    dst = wimgs + (size_t)row * 1024;
  } else {
    int r = row - 128 * 36;
    src = caps + (size_t)r * 1024;
    dst = wcaps + (size_t)r * 1024;
  }
  float v[4];
  float s = 0.f;
#pragma unroll
  for (int e = 0; e < 4; ++e) {
    v[e] = src[threadIdx.x + 256 * e];
    s += v[e] * v[e];
  }
#pragma unroll
  for (int off = 16; off > 0; off >>= 1) s += __shfl_xor(s, off, 32);
  __shared__ float red[8];
  if ((threadIdx.x & 31) == 0) red[threadIdx.x >> 5] = s;
  __syncthreads();
  float tot = 0.f;
#pragma unroll
  for (int k = 0; k < 8; ++k) tot += red[k];
  const float inv = 1.0f / fmaxf(sqrtf(tot), 1e-12f);
#pragma unroll
  for (int e = 0; e < 4; ++e) dst[threadIdx.x + 256 * e] = v[e] * inv;
}

// Compute per-chunk slabs into LDS:
//  X_w[i][d] = sum_l omega2[i][w][l] * caps[tg][l][d0+d]   (M=i, K=l, N=d)
//  Y_w[t][d] = sum_k omega1[w][t][k] * imgs[ig][k][d0+d]   (M=t, K=k, N=d)
static __device__ __forceinline__ void slabs(float* lds,
                                             const float* __restrict__ gi,
                                             const float* __restrict__ gc,
                                             int w, int ln, int koff, int ioff,
                                             int tg, int ig, int d0) {
  v8f x; x = 0.0f;
  for (int l0 = 0; l0 < 60; l0 += 4) {
    v2f a = *(const v2f*)&lds[OM2_O + (ln * 16 + w) * 64 + l0 + koff];
    v2f b;
    b.x = gc[((size_t)tg * 60 + (l0 + koff)) * 1024 + d0 + ln];
    b.y = gc[((size_t)tg * 60 + (l0 + koff + 1)) * 1024 + d0 + ln];
    x = wmma4(a, b, x);
  }
  v8f y; y = 0.0f;
  for (int k0 = 0; k0 < 36; k0 += 4) {
    v2f a = *(const v2f*)&lds[OM1_O + (w * 16 + ln) * 48 + k0 + koff];
    v2f b;
    b.x = gi[((size_t)ig * 36 + (k0 + koff)) * 1024 + d0 + ln];
    b.y = gi[((size_t)ig * 36 + (k0 + koff + 1)) * 1024 + d0 + ln];
    y = wmma4(a, b, y);
  }
#pragma unroll
  for (int j = 0; j < 8; ++j) {
    lds[XS_O + w * 256 + (j + ioff) * 16 + ln] = x[j];
    lds[YS_O + w * 256 + (j + ioff) * 16 + ln] = y[j];
  }
}

// ---------------- kernel 2: one 16i x 16t tile per workgroup ----------------
__global__ __launch_bounds__(512) void nmf_tile(const float* __restrict__ gi,
                                                const float* __restrict__ gc,
                                                const int* __restrict__ il_g,
                                                const int* __restrict__ cl_g,
                                                float* __restrict__ out) {
  extern __shared__ float lds[];
  const int tid = threadIdx.x;
  const int w = tid >> 5;         // wave 0..15
  const int lane = tid & 31;
  const int hi = lane >> 4;       // half-wave
  const int ln = lane & 15;
  const int ib = (blockIdx.x >> 3) * 16;  // image tile base
  const int tb = (blockIdx.x & 7) * 16;   // caption tile base
  const int tg = tb + w;                  // this wave's caption (X / dV side)
  const int ig = ib + w;                  // this wave's image   (Y / dU side)
  const int ioff = hi ? 8 : 0;            // C/D row offset (M = j + ioff)
  const int koff = hi ? 2 : 0;            // A/B K-slot offset

  // ---- init masks and omegas ----
  for (int idx = tid; idx < 16 * 64; idx += 512) {
    int t = idx >> 6, l = idx & 63;
    lds[CM_O + idx] = (l < cl_g[tb + t]) ? 1.f : 0.f;
  }
  for (int idx = tid; idx < 16 * 48; idx += 512) {
    int i = idx / 48, k = idx % 48;
    lds[IM_O + idx] = (k < il_g[ib + i]) ? 1.f : 0.f;
  }
  for (int idx = tid; idx < 16 * 16 * 64; idx += 512) {
    int t = (idx >> 6) & 15, l = idx & 63;
    int cl = cl_g[tb + t];
    lds[OM2_O + idx] = (l < cl) ? (1.f / (float)cl) : 0.f;
  }
  for (int idx = tid; idx < 16 * 16 * 48; idx += 512) {
    int i = idx / (16 * 48), k = idx % 48;
    int il = il_g[ib + i];
    lds[OM1_O + idx] = (k < il) ? (1.f / (float)il) : 0.f;
  }
  __syncthreads();

#pragma unroll 1
  for (int iter = 0; iter < N_ITERS; ++iter) {
    // ============ phase 1: omega_2 -= eta * res_1 @ caps^T ============
    v8f dv[4];
#pragma unroll
    for (int n = 0; n < 4; ++n) dv[n] = 0.0f;
#pragma unroll 1
    for (int ch = 0; ch < 64; ++ch) {
      const int d0 = ch * 16;
      slabs(lds, gi, gc, w, ln, koff, ioff, tg, ig, d0);
      __syncthreads();
      // dv[n] += (X_w - Y_*) @ caps[tg]^T   (M=i, K=d, N=l)
#pragma unroll
      for (int n = 0; n < 4; ++n) {
        const int lc = (n * 16 + ln < 59) ? (n * 16 + ln) : 59;  // pad clamp
#pragma unroll
        for (int s = 0; s < 4; ++s) {
          const int dd = s * 4 + koff;
          v2f ax = *(const v2f*)&lds[XS_O + w * 256 + ln * 16 + dd];
          v2f ay = *(const v2f*)&lds[YS_O + ln * 256 + w * 16 + dd];
          v2f a; a.x = ax.x - ay.x; a.y = ax.y - ay.y;
          v2f b = *(const v2f*)&gc[((size_t)tg * 60 + lc) * 1024 + d0 + dd];
          dv[n] = wmma4(a, b, dv[n]);
        }
      }
      __syncthreads();
    }
    // update + relu + mask (wave w owns t=w rows)
#pragma unroll
    for (int n = 0; n < 4; ++n)
#pragma unroll
      for (int j = 0; j < 8; ++j) {
        const int i = j + ioff, l = n * 16 + ln;
        const int idx = (i * 16 + w) * 64 + l;
        float nv = lds[OM2_O + idx] - ETA_F * dv[n][j];
        lds[OM2_O + idx] = fmaxf(nv, 0.f) * lds[CM_O + w * 64 + l];
      }
    __syncthreads();
    if (tid < 256) {  // pair p = i*16+t == tid
      float s = 0.f;
      for (int l = 0; l < 64; ++l) s += lds[OM2_O + tid * 64 + l];
      lds[SC_O + tid] = 1.f / (s + EPS_F);
    }
    __syncthreads();
    for (int idx = tid; idx < 16384; idx += 512)
      lds[OM2_O + idx] *= lds[SC_O + (idx >> 6)];
    __syncthreads();

    // ============ phase 2: omega_1 += eta * res_2 @ imgs^T ============
    v8f du[3];
#pragma unroll
    for (int n = 0; n < 3; ++n) du[n] = 0.0f;
#pragma unroll 1
    for (int ch = 0; ch < 64; ++ch) {
      const int d0 = ch * 16;
      slabs(lds, gi, gc, w, ln, koff, ioff, tg, ig, d0);
      __syncthreads();
      // du[n] += (X_* - Y_w) @ imgs[ig]^T   (M=t, K=d, N=k)
#pragma unroll
      for (int n = 0; n < 3; ++n) {
        const int kc = (n * 16 + ln < 35) ? (n * 16 + ln) : 35;  // pad clamp
#pragma unroll
        for (int s = 0; s < 4; ++s) {
          const int dd = s * 4 + koff;
          v2f ax = *(const v2f*)&lds[XS_O + ln * 256 + w * 16 + dd];
          v2f ay = *(const v2f*)&lds[YS_O + w * 256 + ln * 16 + dd];
          v2f a; a.x = ax.x - ay.x; a.y = ax.y - ay.y;
          v2f b = *(const v2f*)&gi[((size_t)ig * 36 + kc) * 1024 + d0 + dd];
          du[n] = wmma4(a, b, du[n]);
        }
      }
      __syncthreads();
    }
    // update + relu + mask (wave w owns i=w rows)
#pragma unroll
    for (int n = 0; n < 3; ++n)
#pragma unroll
      for (int j = 0; j < 8; ++j) {
        const int t = j + ioff, k = n * 16 + ln;
        const int idx = (w * 16 + t) * 48 + k;
        float nv = lds[OM1_O + idx] + ETA_F * du[n][j];
        lds[OM1_O + idx] = fmaxf(nv, 0.f) * lds[IM_O + w * 48 + k];
      }
    __syncthreads();
    if (tid < 256) {
      float s = 0.f;
      for (int k = 0; k < 48; ++k) s += lds[OM1_O + tid * 48 + k];
      lds[SC_O + tid] = 1.f / (s + EPS_F);
    }
    __syncthreads();
    for (int idx = tid; idx < 12288; idx += 512)
      lds[OM1_O + idx] *= lds[SC_O + idx / 48];
    __syncthreads();
  }

  // ============ final: sims = <g_img, g_cap> with l2norms ============
  float s11 = 0.f, s22 = 0.f, s12 = 0.f;
#pragma unroll 1
  for (int ch = 0; ch < 64; ++ch) {
    slabs(lds, gi, gc, w, ln, koff, ioff, tg, ig, ch * 16);
    __syncthreads();
    if (tid < 256) {
      const int pi = tid >> 4, pt = tid & 15;
#pragma unroll
      for (int d = 0; d < 16; ++d) {
        float g1 = lds[YS_O + pi * 256 + pt * 16 + d];  // omega1 . imgs
        float g2 = lds[XS_O + pt * 256 + pi * 16 + d];  // omega2 . caps
        s12 += g1 * g2; s11 += g1 * g1; s22 += g2 * g2;
      }
    }
    __syncthreads();
  }
  if (tid < 256) {
    const int pi = tid >> 4, pt = tid & 15;
    const float dnm = fmaxf(sqrtf(s11), 1e-12f) * fmaxf(sqrtf(s22), 1e-12f);
    out[(size_t)(ib + pi) * 128 + (tb + pt)] = s12 / dnm;
  }
}

extern "C" void kernel_launch(void* const* d_in, const int* in_sizes, int n_in,
                              void* d_out, int out_size, void* d_ws, size_t ws_size,
                              hipStream_t stream) {
  (void)in_sizes; (void)n_in; (void)out_size; (void)ws_size;
  const float* imgs = (const float*)d_in[0];   // [128][36][1024] f32
  const float* caps = (const float*)d_in[1];   // [128][60][1024] f32
  const int* il = (const int*)d_in[2];         // [128] i32
  const int* cl = (const int*)d_in[3];         // [128] i32
  // workspace: normalized imgs (18.9MB) + normalized caps (31.5MB) = ~48MB
  float* wimgs = (float*)d_ws;
  float* wcaps = wimgs + (size_t)128 * 36 * 1024;

  l2norm_rows<<<128 * 36 + 128 * 60, 256, 0, stream>>>(imgs, caps, wimgs, wcaps);
  nmf_tile<<<64, 512, LDS_FLOATS * sizeof(float), stream>>>(wimgs, wcaps, il, cl,
                                                            (float*)d_out);
}